// GraphAttentionLayer_33225867002153
// MI455X (gfx1250) — compile-verified
//
#include <hip/hip_runtime.h>
#include <hip/hip_bf16.h>

typedef float v2f __attribute__((ext_vector_type(2)));
typedef float v8f __attribute__((ext_vector_type(8)));

#define DIM 128
// float2-unit row stride for swizzled W in LDS.
// 2*BSTRIDE % 64 == 32  ->  lane halves (kk vs kk+1 rows) hit disjoint LDS bank sets.
#define BSTRIDE 144

__device__ __forceinline__ float lrelu02(float v) { return v > 0.0f ? v : 0.2f * v; }

// order-preserving float<->uint encoding so atomicMax(u32) == float max
__device__ __forceinline__ unsigned f2ord(float f) {
    unsigned u = __float_as_uint(f);
    return (u & 0x80000000u) ? ~u : (u | 0x80000000u);
}
__device__ __forceinline__ float ord2f(unsigned u) {
    return (u & 0x80000000u) ? __uint_as_float(u & 0x7fffffffu) : __uint_as_float(~u);
}

// ---------------------------------------------------------------------------
// Kernel 1: x = input @ W via V_WMMA_F32_16X16X4_F32.
// Block = 256 threads = 8 waves; each wave owns 16 rows x 128 cols.
// W staged in LDS pre-swizzled into K-pair float2s so each WMMA B operand is
// one ds_load_b64 into an even-aligned VGPR pair (no register shuffles).
// ---------------------------------------------------------------------------
__global__ void __launch_bounds__(256) gat_gemm_kernel(const float* __restrict__ input,
                                                       const float* __restrict__ W,
                                                       float* __restrict__ x, int N) {
    __shared__ v2f Wlds[64 * BSTRIDE];  // 72 KB
    // swizzled stage: Wlds[kk*BSTRIDE + c] = (W[2kk][c], W[2kk+1][c])
    for (int idx = threadIdx.x; idx < 64 * DIM; idx += 256) {
        const int kk = idx >> 7;    // K-pair index 0..63
        const int c  = idx & 127;   // column
        v2f p;
        p.x = W[(2 * kk + 0) * DIM + c];
        p.y = W[(2 * kk + 1) * DIM + c];
        Wlds[kk * BSTRIDE + c] = p;
    }
    __syncthreads();

    const int wave = threadIdx.x >> 5;
    const int lane = threadIdx.x & 31;
    const int row0 = (blockIdx.x * 8 + wave) * 16;
    if (row0 >= N) return;  // wave-uniform: EXEC stays all-ones for WMMA

    const int half = lane >> 4;   // 0: K pair {0,1}, 1: K pair {2,3}
    const int l16  = lane & 15;

    // A layout (16x4 f32): lane%16 = M row, VGPR0/1 = K+0/K+1, halves carry K+2/K+3
    int arow = row0 + l16;
    if (arow >= N) arow = N - 1;  // clamp loads (stores guarded below)
    const float* aptr = input + (size_t)arow * DIM + 2 * half;

    v8f acc[8];
#pragma unroll
    for (int t = 0; t < 8; ++t) acc[t] = (v8f){0.f, 0.f, 0.f, 0.f, 0.f, 0.f, 0.f, 0.f};

    for (int k = 0; k < DIM; k += 4) {
        v2f a = *reinterpret_cast<const v2f*>(aptr + k);      // global_load_b64
        const v2f* brow = &Wlds[((k >> 1) + half) * BSTRIDE + l16];
#pragma unroll
        for (int t = 0; t < 8; ++t) {
            v2f b = brow[t * 16];                             // ds_load_b64
            acc[t] = __builtin_amdgcn_wmma_f32_16x16x4_f32(
                false, a, false, b, (short)0, acc[t], false, false);
        }
    }

    // C/D layout: VGPR j -> row j + 8*half, col lane%16
#pragma unroll
    for (int t = 0; t < 8; ++t) {
#pragma unroll
        for (int j = 0; j < 8; ++j) {
            const int r = row0 + j + 8 * half;
            if (r < N) x[(size_t)r * DIM + t * 16 + l16] = acc[t][j];
        }
    }
}

// ---------------------------------------------------------------------------
// Kernel 2: per-node attention scores s_src = x.a_lo, s_dst = x.a_hi
// One wave per node, float4 per lane, full-wave shuffle reduction.
// ---------------------------------------------------------------------------
__global__ void __launch_bounds__(256) gat_scores_kernel(const float* __restrict__ x,
                                                         const float* __restrict__ a,
                                                         float* __restrict__ s_src,
                                                         float* __restrict__ s_dst, int N) {
    const int wave = threadIdx.x >> 5;
    const int lane = threadIdx.x & 31;
    const int node = blockIdx.x * 8 + wave;
    if (node >= N) return;

    float4 xi = reinterpret_cast<const float4*>(x + (size_t)node * DIM)[lane];
    float4 al = reinterpret_cast<const float4*>(a)[lane];
    float4 ah = reinterpret_cast<const float4*>(a + DIM)[lane];

    float ss = xi.x * al.x + xi.y * al.y + xi.z * al.z + xi.w * al.w;
    float sd = xi.x * ah.x + xi.y * ah.y + xi.z * ah.z + xi.w * ah.w;
#pragma unroll
    for (int off = 16; off > 0; off >>= 1) {
        ss += __shfl_down(ss, off);
        sd += __shfl_down(sd, off);
    }
    if (lane == 0) {
        s_src[node] = ss;
        s_dst[node] = sd;
    }
}

// ---------------------------------------------------------------------------
// Kernel 3: init agg accumulator (d_out) to 0, m_enc to 0 (== -inf), denom to 0
// ---------------------------------------------------------------------------
__global__ void __launch_bounds__(256) gat_init_kernel(float4* __restrict__ out4,
                                                       unsigned* __restrict__ m_enc,
                                                       float* __restrict__ denom, int N) {
    const int idx = blockIdx.x * 256 + threadIdx.x;
    const int nvec = N * (DIM / 4);
    if (idx < nvec) out4[idx] = make_float4(0.f, 0.f, 0.f, 0.f);
    if (idx < N) {
        m_enc[idx] = 0u;  // f2ord of most-negative float
        denom[idx] = 0.f;
    }
}

// ---------------------------------------------------------------------------
// Kernel 4: segment max via atomicMax on order-encoded uint
// ---------------------------------------------------------------------------
__global__ void __launch_bounds__(256) gat_edge_max_kernel(const int* __restrict__ triple,
                                                           const float* __restrict__ s_src,
                                                           const float* __restrict__ s_dst,
                                                           unsigned* __restrict__ m_enc, int E) {
    const int e = blockIdx.x * 256 + threadIdx.x;
    if (e >= E) return;
    const int h = triple[3 * e + 0];
    const int t = triple[3 * e + 2];
    const float v = lrelu02(s_src[h] + s_dst[t]);
    atomicMax(&m_enc[h], f2ord(v));
}

// ---------------------------------------------------------------------------
// Kernel 5: ex = exp(e - m[h]); denom[h] += ex (atomic f32)
// ---------------------------------------------------------------------------
__global__ void __launch_bounds__(256) gat_edge_sum_kernel(const int* __restrict__ triple,
                                                           const float* __restrict__ s_src,
                                                           const float* __restrict__ s_dst,
                                                           const unsigned* __restrict__ m_enc,
                                                           float* __restrict__ denom,
                                                           float* __restrict__ ex, int E) {
    const int e = blockIdx.x * 256 + threadIdx.x;
    if (e >= E) return;
    const int h = triple[3 * e + 0];
    const int t = triple[3 * e + 2];
    const float v = lrelu02(s_src[h] + s_dst[t]);
    const float m = ord2f(m_enc[h]);
    const float exv = __expf(v - m);
    ex[e] = exv;
    atomicAdd(&denom[h], exv);
}

// ---------------------------------------------------------------------------
// Kernel 6: agg[h] += (ex/denom[h]) * x[t]   — one wave per edge, float4 lanes
// ---------------------------------------------------------------------------
__global__ void __launch_bounds__(256) gat_edge_agg_kernel(const int* __restrict__ triple,
                                                           const float* __restrict__ ex,
                                                           const float* __restrict__ denom,
                                                           const float* __restrict__ x,
                                                           float* __restrict__ out, int E) {
    const int wave = threadIdx.x >> 5;
    const int lane = threadIdx.x & 31;
    const int e = blockIdx.x * 8 + wave;
    if (e >= E) return;
    const int h = triple[3 * e + 0];
    const int t = triple[3 * e + 2];
    const float att = ex[e] / denom[h];
    const float4 xt = reinterpret_cast<const float4*>(x + (size_t)t * DIM)[lane];
    float* dst = out + (size_t)h * DIM + lane * 4;
    atomicAdd(dst + 0, att * xt.x);
    atomicAdd(dst + 1, att * xt.y);
    atomicAdd(dst + 2, att * xt.z);
    atomicAdd(dst + 3, att * xt.w);
}

// ---------------------------------------------------------------------------
// Kernel 7: out = elu(agg + x), in place over d_out
// ---------------------------------------------------------------------------
__global__ void __launch_bounds__(256) gat_final_kernel(float4* __restrict__ out4,
                                                        const float4* __restrict__ x4, int N) {
    const int idx = blockIdx.x * 256 + threadIdx.x;
    const int nvec = N * (DIM / 4);
    if (idx >= nvec) return;
    float4 o = out4[idx];
    const float4 xv = x4[idx];
    o.x += xv.x; o.y += xv.y; o.z += xv.z; o.w += xv.w;
    o.x = o.x > 0.f ? o.x : __expf(o.x) - 1.f;
    o.y = o.y > 0.f ? o.y : __expf(o.y) - 1.f;
    o.z = o.z > 0.f ? o.z : __expf(o.z) - 1.f;
    o.w = o.w > 0.f ? o.w : __expf(o.w) - 1.f;
    out4[idx] = o;
}

extern "C" void kernel_launch(void* const* d_in, const int* in_sizes, int n_in,
                              void* d_out, int out_size, void* d_ws, size_t ws_size,
                              hipStream_t stream) {
    const float* input = (const float*)d_in[0];   // (N,128) f32
    const float* W     = (const float*)d_in[1];   // (128,128) f32
    const float* a     = (const float*)d_in[2];   // (256,1) f32
    const int*   triple= (const int*)d_in[3];     // (E,3) i32
    float* out = (float*)d_out;

    const int N = in_sizes[0] / DIM;
    const int E = in_sizes[3] / 3;

    // workspace layout (all 16B-aligned at base):
    char* ws = (char*)d_ws;
    float*    x      = (float*)ws;                               // N*128 f32
    float*    s_src  = (float*)(ws + (size_t)N * DIM * 4);       // N f32
    float*    s_dst  = s_src + N;                                // N f32
    unsigned* m_enc  = (unsigned*)(s_dst + N);                   // N u32
    float*    denom  = (float*)(m_enc + N);                      // N f32
    float*    ex     = denom + N;                                // E f32

    gat_gemm_kernel<<<(N + 127) / 128, 256, 0, stream>>>(input, W, x, N);
    gat_scores_kernel<<<(N + 7) / 8, 256, 0, stream>>>(x, a, s_src, s_dst, N);
    gat_init_kernel<<<(N * (DIM / 4) + 255) / 256, 256, 0, stream>>>((float4*)out, m_enc, denom, N);
    gat_edge_max_kernel<<<(E + 255) / 256, 256, 0, stream>>>(triple, s_src, s_dst, m_enc, E);
    gat_edge_sum_kernel<<<(E + 255) / 256, 256, 0, stream>>>(triple, s_src, s_dst, m_enc, denom, ex, E);
    gat_edge_agg_kernel<<<(E + 7) / 8, 256, 0, stream>>>(triple, ex, denom, x, out, E);
    gat_final_kernel<<<(N * (DIM / 4) + 255) / 256, 256, 0, stream>>>((float4*)out, (const float4*)x, N);
}